// Sampler_44040594653444
// MI455X (gfx1250) — compile-verified
//
#include <hip/hip_runtime.h>
#include <stdint.h>
#include <math.h>

// ---------------------------------------------------------------------------
// Top-p sampler for MI455X (gfx1250).
//   rows=128, vocab=128000 (row stride 128256), T=0.7, top_p=0.9
// Bandwidth-bound: ~131MB mandatory HBM traffic (~5.6us @ 23.3TB/s).
// Logits (64MB) fit in 192MB L2 -> multi-pass radix-histogram threshold
// selection instead of a sort. Pass 1 streams through LDS with CDNA5
// async loads (global_load_async_to_lds_b128 + s_wait_asynccnt).
// Block-wide f32 sum reduction for Z uses V_WMMA_F32_16X16X4_F32 with an
// all-ones B matrix (exact f32 FMA chain, 4 matrix ops replace a tree).
// ---------------------------------------------------------------------------

#define ROWS    128
#define VOCAB   128000
#define VSTRIDE 128256
#define TPB     256
#define NCHUNK  (VOCAB / (TPB * 4))   // 125 exact
#define NBIN0   2048                  // key bits [31:21]
#define NBIN1   2048                  // key bits [20:10]
#define NBIN2   1024                  // key bits [ 9: 0]

typedef __attribute__((ext_vector_type(2))) float v2f;
typedef __attribute__((ext_vector_type(8))) float v8f;

__device__ __forceinline__ uint32_t ordkey(float x) {
    uint32_t b = __float_as_uint(x);
    uint32_t m = (uint32_t)((int32_t)b >> 31) | 0x80000000u;
    return b ^ m;   // monotone: a<b (float) <=> ordkey(a)<ordkey(b)
}

__device__ __forceinline__ uint32_t rotl32(uint32_t x, int r) {
    return (x << r) | (x >> (32 - r));
}

// Threefry-2x32, 20 rounds (JAX-style PRNG core).
__device__ __forceinline__ void threefry2x32(uint32_t k0, uint32_t k1,
                                             uint32_t c0, uint32_t c1,
                                             uint32_t& o0, uint32_t& o1) {
    uint32_t ks2 = 0x1BD11BDAu ^ k0 ^ k1;
    uint32_t x0 = c0 + k0, x1 = c1 + k1;
    const int ra[4] = {13, 15, 26, 6};
    const int rb[4] = {17, 29, 16, 24};
#pragma unroll
    for (int i = 0; i < 4; ++i) { x0 += x1; x1 = rotl32(x1, ra[i]); x1 ^= x0; }
    x0 += k1; x1 += ks2 + 1u;
#pragma unroll
    for (int i = 0; i < 4; ++i) { x0 += x1; x1 = rotl32(x1, rb[i]); x1 ^= x0; }
    x0 += ks2; x1 += k0 + 2u;
#pragma unroll
    for (int i = 0; i < 4; ++i) { x0 += x1; x1 = rotl32(x1, ra[i]); x1 ^= x0; }
    x0 += k0; x1 += k1 + 3u;
#pragma unroll
    for (int i = 0; i < 4; ++i) { x0 += x1; x1 = rotl32(x1, rb[i]); x1 ^= x0; }
    x0 += k1; x1 += ks2 + 4u;
#pragma unroll
    for (int i = 0; i < 4; ++i) { x0 += x1; x1 = rotl32(x1, ra[i]); x1 ^= x0; }
    o0 = x0 + ks2; o1 = x1 + k0 + 5u;
}

__global__ __launch_bounds__(TPB) void topp_sampler_kernel(
    const float* __restrict__ logits,   // [128, 128256]
    float* __restrict__ out_tokens,     // [128]   (token index as float)
    float* __restrict__ out_probs)      // [128, 128000]
{
    const int row = blockIdx.x;
    const int tid = threadIdx.x;
    const float* rowp = logits + (size_t)row * VSTRIDE;
    constexpr float INV_T = 1.0f / 0.7f;
    constexpr float TOP_P = 0.9f;

    __shared__ float s_stage[2][TPB * 4];   // async DMA double buffer (8KB)
    __shared__ float s_hist[NBIN0];         // radix histogram (8KB)
    __shared__ float s_red[TPB];
    __shared__ int   s_redi[TPB];
    __shared__ float s_aMax, s_cum, s_Zk;
    __shared__ int   s_b0, s_b1;
    __shared__ uint32_t s_K;

    // ---------------- Pass 1: row max via async global->LDS streaming ------
    {
        const uint64_t sbase = (uint64_t)(uintptr_t)rowp;
        const uint32_t lds0 = (uint32_t)(uintptr_t)&s_stage[0][0];
        const uint32_t lds1 = (uint32_t)(uintptr_t)&s_stage[1][0];
        const uint32_t myByte = (uint32_t)tid * 16u;

        // prime chunk 0
        asm volatile("global_load_async_to_lds_b128 %0, %1, %2"
                     :: "v"(lds0 + myByte), "v"(myByte), "s"(sbase)
                     : "memory");

        float m = -INFINITY;
        for (int ch = 0; ch < NCHUNK; ++ch) {
            if (ch + 1 < NCHUNK) {
                uint32_t nxt = (uint32_t)(ch + 1) * (TPB * 16u) + myByte;
                uint32_t dst = (((ch + 1) & 1) ? lds1 : lds0) + myByte;
                asm volatile("global_load_async_to_lds_b128 %0, %1, %2"
                             :: "v"(dst), "v"(nxt), "s"(sbase)
                             : "memory");
                asm volatile("s_wait_asynccnt 1" ::: "memory");
            } else {
                asm volatile("s_wait_asynccnt 0" ::: "memory");
            }
            // each thread consumes exactly what its own wave's lane DMA'd
            const float4 v = *(const float4*)&s_stage[ch & 1][tid * 4];
            m = fmaxf(m, fmaxf(fmaxf(v.x, v.y), fmaxf(v.z, v.w)));
        }
        s_red[tid] = m;
        __syncthreads();
        for (int s = TPB / 2; s > 0; s >>= 1) {
            if (tid < s) s_red[tid] = fmaxf(s_red[tid], s_red[tid + s]);
            __syncthreads();
        }
        if (tid == 0) s_aMax = s_red[0] * INV_T;
        __syncthreads();
    }
    const float aMax = s_aMax;

    // ---------------- Pass 2: Z + level-0 histogram (L2-resident reads) ----
    for (int i = tid; i < NBIN0; i += TPB) s_hist[i] = 0.0f;
    __syncthreads();
    float Z;
    {
        float zp = 0.0f;
        for (int ch = 0; ch < NCHUNK; ++ch) {
            const float4 v = *(const float4*)(rowp + ch * (TPB * 4) + tid * 4);
            const float xs[4] = {v.x, v.y, v.z, v.w};
#pragma unroll
            for (int j = 0; j < 4; ++j) {
                float e = __expf(xs[j] * INV_T - aMax);
                zp += e;
                atomicAdd(&s_hist[ordkey(xs[j]) >> 21], e);   // ds_add_f32
            }
        }
        s_red[tid] = zp;
        __syncthreads();

        // Block-wide f32 sum of 256 partials via 4 chained WMMAs:
        //   D = A(16x4) x ones(4x16) + C  => D[m][n] = sum_k A[m][k]
        // A[m][k] = s_red[i*64 + m*4 + k]; ISA layout: lanes 0-15 hold
        // K=0..1, lanes 16-31 hold K=2..3. Every wave runs this redundantly
        // on the same LDS data so EXEC is all-1s and Z lands in-register.
        const int lane = tid & 31;
        const int m16  = lane & 15;
        const int hk   = (lane >> 4) * 2;     // 0 or 2
        const v2f ones = {1.0f, 1.0f};
        v8f c = {};
#pragma unroll
        for (int i = 0; i < 4; ++i) {
            v2f a;
            a.x = s_red[i * 64 + m16 * 4 + hk + 0];
            a.y = s_red[i * 64 + m16 * 4 + hk + 1];
            c = __builtin_amdgcn_wmma_f32_16x16x4_f32(
                    false, a, false, ones, (short)0, c, false, false);
        }
        float s = c[0] + c[1] + c[2] + c[3] + c[4] + c[5] + c[6] + c[7];
        Z = s + __shfl_xor(s, 16);            // rows 0-7 half + rows 8-15 half
        __syncthreads();
    }
    const float target = TOP_P * Z;

    // level-0 top-down scan
    if (tid == 0) {
        float cum = 0.0f; int b = NBIN0 - 1;
        for (; b > 0; --b) {
            float h = s_hist[b];
            if (cum + h >= target) break;
            cum += h;
        }
        s_b0 = b; s_cum = cum;
    }
    __syncthreads();

    // ---------------- Pass 3: level-1 histogram within bin b0 --------------
    {
        const uint32_t b0 = (uint32_t)s_b0;
        __syncthreads();
        for (int i = tid; i < NBIN1; i += TPB) s_hist[i] = 0.0f;
        __syncthreads();
        for (int ch = 0; ch < NCHUNK; ++ch) {
            const float4 v = *(const float4*)(rowp + ch * (TPB * 4) + tid * 4);
            const float xs[4] = {v.x, v.y, v.z, v.w};
#pragma unroll
            for (int j = 0; j < 4; ++j) {
                uint32_t k = ordkey(xs[j]);
                if ((k >> 21) == b0)
                    atomicAdd(&s_hist[(k >> 10) & 0x7FFu],
                              __expf(xs[j] * INV_T - aMax));
            }
        }
        __syncthreads();
        if (tid == 0) {
            float cum = s_cum; int b = NBIN1 - 1;
            for (; b > 0; --b) {
                float h = s_hist[b];
                if (cum + h >= target) break;
                cum += h;
            }
            s_b1 = b; s_cum = cum;
        }
        __syncthreads();
    }

    // ---------------- Pass 4: level-2 histogram -> exact 32-bit key --------
    {
        const uint32_t pre = ((uint32_t)s_b0 << 11) | (uint32_t)s_b1;
        __syncthreads();
        for (int i = tid; i < NBIN2; i += TPB) s_hist[i] = 0.0f;
        __syncthreads();
        for (int ch = 0; ch < NCHUNK; ++ch) {
            const float4 v = *(const float4*)(rowp + ch * (TPB * 4) + tid * 4);
            const float xs[4] = {v.x, v.y, v.z, v.w};
#pragma unroll
            for (int j = 0; j < 4; ++j) {
                uint32_t k = ordkey(xs[j]);
                if ((k >> 10) == pre)
                    atomicAdd(&s_hist[k & 0x3FFu],
                              __expf(xs[j] * INV_T - aMax));
            }
        }
        __syncthreads();
        if (tid == 0) {
            float cum = s_cum; int b = NBIN2 - 1;
            for (; b > 0; --b) {
                float h = s_hist[b];
                if (cum + h >= target) break;
                cum += h;
            }
            s_K  = (pre << 10) | (uint32_t)b;       // threshold key
            s_Zk = cum + s_hist[b];                 // kept mass (ties kept)
        }
        __syncthreads();
    }
    const uint32_t K   = s_K;
    const float invZk  = 1.0f / s_Zk;
    const float logZk  = __logf(s_Zk);

    // ---------------- Pass 5: write probs + fused Gumbel-argmax sample -----
    {
        // skey = threefry(key(0)=[0,0], [0,1])  (fold_in-style derivation)
        uint32_t sk0, sk1;
        threefry2x32(0u, 0u, 0u, 1u, sk0, sk1);

        float best = -INFINITY; int bestIdx = 0;
        float* prow = out_probs + (size_t)row * VOCAB;
        for (int ch = 0; ch < NCHUNK; ++ch) {
            const int cbase = ch * (TPB * 4) + tid * 4;
            const float4 v = *(const float4*)(rowp + cbase);
            const float xs[4] = {v.x, v.y, v.z, v.w};
            float4 pv;
            float* pp = (float*)&pv;
#pragma unroll
            for (int j = 0; j < 4; ++j) {
                const int c = cbase + j;
                const float a = xs[j] * INV_T - aMax;   // log-prob + logZk
                const bool keep = (ordkey(xs[j]) >= K);
                const float p = keep ? __expf(a) * invZk : 0.0f;
                pp[j] = p;
                if (keep) {
                    uint32_t r0, r1;
                    threefry2x32(sk0, sk1, (uint32_t)row,
                                 (uint32_t)(row * VOCAB + c), r0, r1);
                    float u01 = (float)(r0 >> 8) * 5.9604645e-8f
                                + 2.9802322e-8f;        // (0,1)
                    float g = -__logf(-__logf(u01));
                    float score = (a - logZk) + g;
                    if (score > best) { best = score; bestIdx = c; }
                }
            }
            *(float4*)(prow + cbase) = pv;
        }
        s_red[tid] = best; s_redi[tid] = bestIdx;
        __syncthreads();
        for (int s = TPB / 2; s > 0; s >>= 1) {
            if (tid < s) {
                float o = s_red[tid + s];
                int oi = s_redi[tid + s];
                if (o > s_red[tid] || (o == s_red[tid] && oi < s_redi[tid])) {
                    s_red[tid] = o; s_redi[tid] = oi;
                }
            }
            __syncthreads();
        }
        if (tid == 0) out_tokens[row] = (float)s_redi[0];
    }
}

extern "C" void kernel_launch(void* const* d_in, const int* in_sizes, int n_in,
                              void* d_out, int out_size, void* d_ws, size_t ws_size,
                              hipStream_t stream) {
    (void)in_sizes; (void)n_in; (void)d_ws; (void)ws_size; (void)out_size;
    const float* logits = (const float*)d_in[0];
    float* out = (float*)d_out;            // [128 tokens][128*128000 probs]
    topp_sampler_kernel<<<dim3(ROWS), dim3(TPB), 0, stream>>>(
        logits, out, out + ROWS);
}